// IGCN_57887569215730
// MI455X (gfx1250) — compile-verified
//
#include <hip/hip_runtime.h>

typedef __attribute__((ext_vector_type(2))) float v2f;
typedef __attribute__((ext_vector_type(8))) float v8f;

#define EMB 64

// ---------------- zero fill (float4 stores) ----------------
__global__ __launch_bounds__(256) void zero_f4(float4* __restrict__ p, long long n4) {
    long long i = (long long)blockIdx.x * 256 + threadIdx.x;
    if (i < n4) p[i] = make_float4(0.f, 0.f, 0.f, 0.f);
}

// ---------------- SpMM: y[row] += val * x[col], 64 features ----------------
// 16 threads per edge, float4 per thread. Atomic f32 adds resolve in L2
// (dense operands are 64MB, L2-resident on the 192MB L2).
__global__ __launch_bounds__(256) void spmm_scatter(
    const int* __restrict__ row, const int* __restrict__ col,
    const float* __restrict__ val, const float* __restrict__ x,
    float* __restrict__ y, int nnz) {
    long long gid = (long long)blockIdx.x * 256 + threadIdx.x;
    int e = (int)(gid >> 4);
    if (e >= nnz) return;
    int f = ((int)gid & 15) << 2;
    int r = row[e];
    int c = col[e];
    float v = val[e];
    const float4 xv = *(const float4*)(x + (long long)c * EMB + f);
    float* yp = y + (long long)r * EMB + f;
    atomicAdd(yp + 0, v * xv.x);
    atomicAdd(yp + 1, v * xv.y);
    atomicAdd(yp + 2, v * xv.z);
    atomicAdd(yp + 3, v * xv.w);
}

// ---------------- rep *= w (per-feature); acc = rep ----------------
__global__ __launch_bounds__(256) void scale_w_copy(
    float* __restrict__ rep, float* __restrict__ acc,
    const float* __restrict__ w, long long n) {
    long long i = (long long)blockIdx.x * 256 + threadIdx.x;
    if (i < n) {
        float v = rep[i] * w[i & (EMB - 1)];
        rep[i] = v;
        acc[i] = v;
    }
}

// ---------------- acc += rep ----------------
__global__ __launch_bounds__(256) void axpy_acc(
    float* __restrict__ acc, const float* __restrict__ rep, long long n4) {
    long long i = (long long)blockIdx.x * 256 + threadIdx.x;
    if (i < n4) {
        float4 a = ((float4*)acc)[i];
        float4 r = ((const float4*)rep)[i];
        a.x += r.x; a.y += r.y; a.z += r.z; a.w += r.w;
        ((float4*)acc)[i] = a;
    }
}

// ---------------- Final GEMM: out[b, n] = (acc[users[b]]/4) . (acc[N_USERS+n]/4) ----------------
// One wave per 16x16 output tile, V_WMMA_F32_16X16X4_F32, 16 K-steps (K=64).
// A-matrix 16x4 f32 layout (ISA 7.12.2): lane l -> M=l%16, VGPR pair holds
// K = 2*(l/16)+0 and +1. B symmetric with N=l%16. C/D: VGPR j -> M=j+8*(l/16), N=l%16.
__global__ __launch_bounds__(128) void gemm_wmma_f32(
    const float* __restrict__ acc, const int* __restrict__ users,
    float* __restrict__ out, int n_users, int n_items) {
    const int lane = threadIdx.x & 31;
    const int wave = threadIdx.x >> 5;
    const int tile_n = blockIdx.x * 4 + wave;   // item tile
    const int tile_m = blockIdx.y;              // user tile
    if (tile_n * 16 >= n_items) return;         // wave-uniform: EXEC stays full

    const int half = lane >> 4;                 // 0 or 1
    const int kb = half * 2;                    // K sub-offset for this half-wave
    const int mr = lane & 15;

    const int u = users[tile_m * 16 + mr];
    const float* __restrict__ Arow = acc + (long long)u * EMB;
    const float* __restrict__ Brow = acc + (long long)(n_users + tile_n * 16 + mr) * EMB;

    v8f c = {};
#pragma unroll
    for (int k = 0; k < EMB; k += 4) {
        v2f a = *(const v2f*)(Arow + k + kb);   // 8B-aligned: k%4==0, kb in {0,2}
        v2f b = *(const v2f*)(Brow + k + kb);
        c = __builtin_amdgcn_wmma_f32_16x16x4_f32(
            /*neg_a=*/false, a, /*neg_b=*/false, b,
            /*c_mod=*/(short)0, c, /*reuse_a=*/false, /*reuse_b=*/false);
    }

    const float scale = 0.0625f;                // (1/4)*(1/4) for final_rep = acc/4
    const long long n0 = (long long)tile_n * 16 + mr;
#pragma unroll
    for (int j = 0; j < 8; ++j) {
        int m = j + 8 * half;
        out[(long long)(tile_m * 16 + m) * n_items + n0] = c[j] * scale;
    }
}

extern "C" void kernel_launch(void* const* d_in, const int* in_sizes, int n_in,
                              void* d_out, int out_size, void* d_ws, size_t ws_size,
                              hipStream_t stream) {
    const int*   users    = (const int*)  d_in[0];
    const float* emb      = (const float*)d_in[1];
    const float* w        = (const float*)d_in[2];
    const int*   adj_row  = (const int*)  d_in[3];
    const int*   adj_col  = (const int*)  d_in[4];
    const float* adj_val  = (const float*)d_in[5];
    const int*   feat_row = (const int*)  d_in[6];
    const int*   feat_col = (const int*)  d_in[7];
    const float* feat_val = (const float*)d_in[8];

    const int batch    = in_sizes[0];                 // 512
    const int nnz_adj  = in_sizes[3];                 // 4,000,000
    const int nnz_feat = in_sizes[6];                 // 4,250,000
    const int N        = in_sizes[1] / EMB - 2;       // 250,000 (N_FEAT = N+2)
    const int n_items  = out_size / batch;            // 150,000
    const int n_users  = N - n_items;                 // 100,000

    const long long rowsz = (long long)N * EMB;       // floats per dense matrix
    float* rep0 = (float*)d_ws;
    float* rep1 = rep0 + rowsz;
    float* accb = rep1 + rowsz;

    const long long n4 = rowsz / 4;
    const int zgrid = (int)((n4 + 255) / 256);
    const int sgrid_feat = (int)(((long long)nnz_feat * 16 + 255) / 256);
    const int sgrid_adj  = (int)(((long long)nnz_adj  * 16 + 255) / 256);
    const int agrid = (int)((rowsz + 255) / 256);

    // rep0 = SpMM(feat, emb); rep0 *= w; acc = rep0
    zero_f4<<<zgrid, 256, 0, stream>>>((float4*)rep0, n4);
    spmm_scatter<<<sgrid_feat, 256, 0, stream>>>(feat_row, feat_col, feat_val, emb, rep0, nnz_feat);
    scale_w_copy<<<agrid, 256, 0, stream>>>(rep0, accb, w, rowsz);

    // layer 1: rep1 = SpMM(adj, rep0); acc += rep1
    zero_f4<<<zgrid, 256, 0, stream>>>((float4*)rep1, n4);
    spmm_scatter<<<sgrid_adj, 256, 0, stream>>>(adj_row, adj_col, adj_val, rep0, rep1, nnz_adj);
    axpy_acc<<<zgrid, 256, 0, stream>>>(accb, rep1, n4);

    // layer 2: rep0 = SpMM(adj, rep1); acc += rep0
    zero_f4<<<zgrid, 256, 0, stream>>>((float4*)rep0, n4);
    spmm_scatter<<<sgrid_adj, 256, 0, stream>>>(adj_row, adj_col, adj_val, rep1, rep0, nnz_adj);
    axpy_acc<<<zgrid, 256, 0, stream>>>(accb, rep0, n4);

    // layer 3: rep1 = SpMM(adj, rep0); acc += rep1
    zero_f4<<<zgrid, 256, 0, stream>>>((float4*)rep1, n4);
    spmm_scatter<<<sgrid_adj, 256, 0, stream>>>(adj_row, adj_col, adj_val, rep0, rep1, nnz_adj);
    axpy_acc<<<zgrid, 256, 0, stream>>>(accb, rep1, n4);

    // out = (acc[users]/4) @ (acc[items]/4)^T   via V_WMMA_F32_16X16X4_F32
    const int tiles_m = (batch + 15) / 16;            // 32
    const int tiles_n = (n_items + 15) / 16;          // 9375
    dim3 ggrid((tiles_n + 3) / 4, tiles_m);
    gemm_wmma_f32<<<ggrid, 128, 0, stream>>>(accb, users, (float*)d_out, n_users, n_items);
}